// SemanticRearrangementModule_61074434949933
// MI455X (gfx1250) — compile-verified
//
#include <hip/hip_runtime.h>
#include <hip/hip_bf16.h>
#include <math.h>

#define EPSF 1e-07f
#define KCLS 19
#define KPAD 20           // bank-conflict padding / zero-pad column for k
#define MROW 32           // padded M rows for the 2x 16-row WMMA tiles
#define CCH  64
#define STRIP 2048        // pixels per workgroup strip

typedef __attribute__((ext_vector_type(2))) float v2f;
typedef __attribute__((ext_vector_type(4))) float f4;
typedef __attribute__((ext_vector_type(4))) int   i4;
typedef __attribute__((ext_vector_type(8))) float v8f;

// ---------------------------------------------------------------------------
// Kernel 0: zero the accumulator region of the workspace (sums, sumsq, counts)
// ---------------------------------------------------------------------------
__global__ void zero_kernel(float* __restrict__ p, int n) {
    int i = blockIdx.x * 256 + threadIdx.x;
    if (i < n) p[i] = 0.0f;
}

// ---------------------------------------------------------------------------
// Kernel 1: per-(b,k,c) sum and sum-of-squares + per-(b,k) counts.
// One strip of STRIP pixels per workgroup; labels cached in LDS; LDS float
// atomics into a padded [c][k] layout (stride KPAD=20 -> lanes sharing a
// channel but differing in class hit distinct banks); global atomic flush.
// ---------------------------------------------------------------------------
__global__ void __launch_bounds__(256)
stats_kernel(const float* __restrict__ x, const int* __restrict__ gt,
             float* __restrict__ gsum, float* __restrict__ gsumsq,
             float* __restrict__ gcnt, int HW) {
    __shared__ float sAcc [CCH * KPAD];   // sum   [c][k]
    __shared__ float sAcc2[CCH * KPAD];   // sumsq [c][k]
    __shared__ float sHist[KCLS];
    __shared__ int   sLab [STRIP];

    const int b      = blockIdx.y;
    const int strip0 = blockIdx.x * STRIP;
    const int tid    = threadIdx.x;

    for (int i = tid; i < CCH * KPAD; i += 256) { sAcc[i] = 0.0f; sAcc2[i] = 0.0f; }
    if (tid < KCLS) sHist[tid] = 0.0f;

    // vectorized label strip load (strip0 is a multiple of 2048 -> 16B aligned)
    const int* gtb = gt + (size_t)b * HW + strip0;
    for (int i = tid; i < STRIP / 4; i += 256)
        ((i4*)sLab)[i] = ((const i4*)gtb)[i];
    __syncthreads();

    // per-strip class histogram (counts)
    for (int i = tid; i < STRIP; i += 256)
        atomicAdd(&sHist[sLab[i]], 1.0f);

    // main accumulation: wave w owns channels w, w+8, ... (coalesced b128 loads)
    const int wave = tid >> 5, lane = tid & 31;
    for (int c = wave; c < CCH; c += 8) {
        const float* xp = x + ((size_t)b * CCH + c) * HW + strip0;
        const int base = c * KPAD;
        for (int it = 0; it < STRIP / 128; ++it) {
            int pix = it * 128 + lane * 4;
            f4 v = __builtin_nontemporal_load((const f4*)(xp + pix));
            i4 lb = *((const i4*)(sLab + pix));
            atomicAdd(&sAcc [base + lb.x], v.x);
            atomicAdd(&sAcc2[base + lb.x], v.x * v.x);
            atomicAdd(&sAcc [base + lb.y], v.y);
            atomicAdd(&sAcc2[base + lb.y], v.y * v.y);
            atomicAdd(&sAcc [base + lb.z], v.z);
            atomicAdd(&sAcc2[base + lb.z], v.z * v.z);
            atomicAdd(&sAcc [base + lb.w], v.w);
            atomicAdd(&sAcc2[base + lb.w], v.w * v.w);
        }
    }
    __syncthreads();

    // flush partials: global layout [b][c][k]
    float* gs = gsum   + (size_t)b * CCH * KCLS;
    float* gq = gsumsq + (size_t)b * CCH * KCLS;
    for (int i = tid; i < CCH * KCLS; i += 256) {
        int c = i / KCLS, k = i % KCLS;
        atomicAdd(&gs[i], sAcc [c * KPAD + k]);
        atomicAdd(&gq[i], sAcc2[c * KPAD + k]);
    }
    if (tid < KCLS) atomicAdd(&gcnt[b * KCLS + tid], sHist[tid]);
}

// ---------------------------------------------------------------------------
// Kernel 2: one block per batch. mean/std from sums; Dirichlet-style weight
// renormalization; mix einsum via V_WMMA_F32_16X16X4_F32 (19x19 @ 19x64,
// zero-padded to 32x20 @ 20x64: 2x4 output tiles, 5 k-steps). All operand
// arrays are zero-padded in LDS so every fragment load is unconditional
// (no exec-mask divergence around the WMMA section). Folds results into
// per-(k,c) affine coefficients A = mix_std/std, B = mix_mean - mean*A.
// ---------------------------------------------------------------------------
__global__ void __launch_bounds__(256)
mix_kernel(const float* __restrict__ gsum, const float* __restrict__ gsumsq,
           const float* __restrict__ gcnt, const float* __restrict__ aug,
           float* __restrict__ coefA, float* __restrict__ coefB) {
    __shared__ float sMean [KPAD * CCH];   // [k][c], row 19 zeroed
    __shared__ float sStd  [KPAD * CCH];   // [k][c], row 19 zeroed
    __shared__ float sW    [MROW * KPAD];  // [row][k], rows>=19 and col 19 zeroed
    __shared__ float sValid[KCLS];

    const int b = blockIdx.x;
    const int tid = threadIdx.x;

    // zero-pad the WMMA operand arrays (covers pad rows/cols)
    for (int i = tid; i < MROW * KPAD; i += 256) sW[i] = 0.0f;
    for (int i = tid; i < CCH; i += 256) {
        sMean[KCLS * CCH + i] = 0.0f;   // k = 19 pad row
        sStd [KCLS * CCH + i] = 0.0f;
    }

    // mean / std (E[x^2] identity)
    for (int i = tid; i < KCLS * CCH; i += 256) {
        int k = i / CCH, c = i % CCH;
        float cnt = gcnt[b * KCLS + k];
        float cs  = cnt > 0.0f ? cnt : 1.0f;
        float su  = gsum  [((size_t)b * CCH + c) * KCLS + k];
        float sq  = gsumsq[((size_t)b * CCH + c) * KCLS + k];
        float mu  = su / cs;
        float vr  = sq / cs - mu * mu;
        vr = vr > 0.0f ? vr : 0.0f;
        sMean[k * CCH + c] = mu;
        sStd [k * CCH + c] = sqrtf(vr) + EPSF;
        if (c == 0) sValid[k] = cnt > 0.0f ? 1.0f : 0.0f;
    }
    __syncthreads();

    // weight rows: w = aug * valid, renormalized to the simplex
    if (tid < KCLS) {
        const float* ar = aug + ((size_t)b * KCLS + tid) * KCLS;
        float tmp[KCLS];
        float s = 0.0f;
        for (int k = 0; k < KCLS; ++k) { tmp[k] = ar[k] * sValid[k]; s += tmp[k]; }
        float d = s > EPSF ? s : EPSF;
        for (int k = 0; k < KCLS; ++k) sW[tid * KPAD + k] = tmp[k] / d;
    }
    __syncthreads();

    // 8 waves -> (Mtile m in {0,1}) x (Ntile n in {0..3})
    const int wave = tid >> 5, lane = tid & 31;
    const int m = wave >> 2, n = wave & 3;
    const int half = lane >> 4, li = lane & 15;
    const int col  = 16 * n + li;

    v8f accM = {};
    v8f accS = {};

#if __has_builtin(__builtin_amdgcn_wmma_f32_16x16x4_f32)
    const int arow = 16 * m + li;          // A: M = lane%16 in both halves
    for (int kk = 0; kk < 5; ++kk) {
        int k0 = 4 * kk + 2 * half;        // A/B: K split across lane halves
        // A fragment (16x4 of W) — unconditional (operands zero-padded)
        v2f a;
        a.x = sW[arow * KPAD + k0];
        a.y = sW[arow * KPAD + k0 + 1];
        // B fragments (4x16): rows striped across lanes ({0,2}/{1,3} per VGPR)
        v2f bm, bs;
        bm.x = sMean[k0 * CCH + col];
        bm.y = sMean[(k0 + 1) * CCH + col];
        bs.x = sStd [k0 * CCH + col];
        bs.y = sStd [(k0 + 1) * CCH + col];
        accM = __builtin_amdgcn_wmma_f32_16x16x4_f32(false, a, false, bm,
                                                     (short)0, accM, false, false);
        accS = __builtin_amdgcn_wmma_f32_16x16x4_f32(false, a, false, bs,
                                                     (short)0, accS, false, false);
    }
#else
    // scalar fallback computing the same D-fragment layout
    for (int i = 0; i < 8; ++i) {
        int row = 16 * m + i + 8 * half;
        float am = 0.0f, as = 0.0f;
        if (row < KCLS) {
            for (int k = 0; k < KCLS; ++k) {
                float w = sW[row * KPAD + k];
                am += w * sMean[k * CCH + col];
                as += w * sStd [k * CCH + col];
            }
        }
        accM[i] = am;
        accS[i] = as;
    }
#endif

    // fold into affine coefficients, write global layout [b][c][k]
    for (int i = 0; i < 8; ++i) {
        int row = 16 * m + i + 8 * half;   // D layout: VGPR i -> M = i (+8 upper half)
        if (row < KCLS) {
            float s  = sStd [row * CCH + col];
            float mu = sMean[row * CCH + col];
            float A  = accS[i] / s;
            float Bc = accM[i] - mu * A;
            coefA[((size_t)b * CCH + col) * KCLS + row] = A;
            coefB[((size_t)b * CCH + col) * KCLS + row] = Bc;
        }
    }
}

// ---------------------------------------------------------------------------
// Kernel 3: streaming apply. out = x*A[gt,c] + B[gt,c], single FMA/element,
// b128 nontemporal load/store, labels + coefficients in LDS.
// ---------------------------------------------------------------------------
__global__ void __launch_bounds__(256)
apply_kernel(const float* __restrict__ x, const int* __restrict__ gt,
             const float* __restrict__ coefA, const float* __restrict__ coefB,
             float* __restrict__ out, int HW) {
    __shared__ float sA[CCH * KPAD];
    __shared__ float sB[CCH * KPAD];
    __shared__ int   sLab[STRIP];

    const int b      = blockIdx.y;
    const int strip0 = blockIdx.x * STRIP;
    const int tid    = threadIdx.x;

    for (int i = tid; i < CCH * KCLS; i += 256) {
        int c = i / KCLS, k = i % KCLS;
        sA[c * KPAD + k] = coefA[(size_t)b * CCH * KCLS + i];
        sB[c * KPAD + k] = coefB[(size_t)b * CCH * KCLS + i];
    }
    const int* gtb = gt + (size_t)b * HW + strip0;
    for (int i = tid; i < STRIP / 4; i += 256)
        ((i4*)sLab)[i] = ((const i4*)gtb)[i];
    __syncthreads();

    const int wave = tid >> 5, lane = tid & 31;
    for (int c = wave; c < CCH; c += 8) {
        const float* xp = x   + ((size_t)b * CCH + c) * HW + strip0;
        float*       op = out + ((size_t)b * CCH + c) * HW + strip0;
        const int base = c * KPAD;
        for (int it = 0; it < STRIP / 128; ++it) {
            int pix = it * 128 + lane * 4;
            f4 v  = __builtin_nontemporal_load((const f4*)(xp + pix));
            i4 lb = *((const i4*)(sLab + pix));
            f4 r;
            r.x = fmaf(v.x, sA[base + lb.x], sB[base + lb.x]);
            r.y = fmaf(v.y, sA[base + lb.y], sB[base + lb.y]);
            r.z = fmaf(v.z, sA[base + lb.z], sB[base + lb.z]);
            r.w = fmaf(v.w, sA[base + lb.w], sB[base + lb.w]);
            __builtin_nontemporal_store(r, (f4*)(op + pix));
        }
    }
}

// ---------------------------------------------------------------------------
extern "C" void kernel_launch(void* const* d_in, const int* in_sizes, int n_in,
                              void* d_out, int out_size, void* d_ws, size_t ws_size,
                              hipStream_t stream) {
    const float* x   = (const float*)d_in[0];
    const int*   gt  = (const int*)  d_in[1];
    const float* aug = (const float*)d_in[2];
    float*       out = (float*)d_out;

    const int B = 4, C = CCH, K = KCLS;
    const int HW = in_sizes[1] / B;          // 512*512

    // workspace layout (floats)
    float* ws     = (float*)d_ws;
    float* gsum   = ws;                       // B*C*K
    float* gsumsq = gsum   + B * C * K;       // B*C*K
    float* gcnt   = gsumsq + B * C * K;       // B*K
    float* cA     = gcnt   + B * K;           // B*C*K
    float* cB     = cA     + B * C * K;       // B*C*K

    const int nz = 2 * B * C * K + B * K;     // accumulators to clear each call
    zero_kernel<<<(nz + 255) / 256, 256, 0, stream>>>(ws, nz);

    dim3 grid(HW / STRIP, B);                 // 128 x 4 strips
    stats_kernel<<<grid, 256, 0, stream>>>(x, gt, gsum, gsumsq, gcnt, HW);
    mix_kernel<<<B, 256, 0, stream>>>(gsum, gsumsq, gcnt, aug, cA, cB);
    apply_kernel<<<grid, 256, 0, stream>>>(x, gt, cA, cB, out, HW);
}